// ModifiedGCN_81939386073613
// MI455X (gfx1250) — compile-verified
//
#include <hip/hip_runtime.h>
#include <hip/hip_bf16.h>

#define NN 100000
#define NE 1600000
#define IN_DIM 128
#define HID 64
#define C2 8
#define GDIM 24
#define NG 64
#define SLOPE 0.01f

typedef float v2f __attribute__((ext_vector_type(2)));
typedef float v8f __attribute__((ext_vector_type(8)));

__device__ __forceinline__ float lrelu(float v) { return v > 0.0f ? v : SLOPE * v; }

// ---------------- fill ----------------
__global__ void fill_kernel(float* __restrict__ p, float v, int n) {
    int i = blockIdx.x * blockDim.x + threadIdx.x;
    if (i < n) p[i] = v;
}

// ---------------- degree ----------------
__global__ void deg_accum_kernel(const int* __restrict__ dst, float* __restrict__ deg, int nE) {
    int e = blockIdx.x * blockDim.x + threadIdx.x;
    if (e < nE) unsafeAtomicAdd(&deg[dst[e]], 1.0f);
}

__global__ void dinv_kernel(float* __restrict__ deg, int n) {
    int i = blockIdx.x * blockDim.x + threadIdx.x;
    if (i < n) deg[i] = rsqrtf(deg[i]);   // deg >= 1 always (self loop)
}

// ---------------- GEMM1: h1[NN,64] = x[NN,128] @ W1[128,64] (WMMA f32 16x16x4) ------
// One wave computes one 16-row tile, 4 column tiles of 16.
// A layout (16x4 f32): lanes 0-15 rows M, v0=K0 v1=K1; lanes 16-31 v0=K2 v1=K3.
// B layout (4x16 f32): v0: K0(lanes0-15)/K2(lanes16-31); v1: K1/K3, N = lane&15.
// C/D layout: VGPRj: M=j (lanes0-15) / M=j+8 (lanes16-31), N = lane&15.
// W is staged in LDS K-pair-interleaved: Ws[p*2*HID + col*2 + t] = W1[(2p+t)*HID + col]
// so every B operand is one aligned ds_load_b64 (no repack movs).
__global__ void gemm1_wmma_kernel(const float* __restrict__ x, const float* __restrict__ W1,
                                  float* __restrict__ h1) {
    __shared__ float Ws[IN_DIM * HID];   // 32 KB, pair-interleaved
    for (int i = threadIdx.x; i < IN_DIM * HID; i += blockDim.x) {
        int p   = i >> 7;          // K-pair index (0..63)
        int rem = i & 127;
        int col = rem >> 1;        // 0..63
        int t   = rem & 1;         // element of pair
        Ws[i] = W1[(size_t)(2 * p + t) * HID + col];
    }
    __syncthreads();

    int wave = threadIdx.x >> 5;
    int lane = threadIdx.x & 31;
    int tile = blockIdx.x * (blockDim.x >> 5) + wave;
    int base = tile * 16;
    if (base >= NN) return;

    int rowl = lane & 15;
    int kh   = lane >> 4;                // 0 or 1
    const float* xrow = x + (size_t)(base + rowl) * IN_DIM;

    v8f acc0 = {}, acc1 = {}, acc2 = {}, acc3 = {};
    for (int kk = 0; kk < IN_DIM; kk += 4) {
        int kbase = kk + 2 * kh;
        int p = kbase >> 1;              // K-pair
        v2f a = *(const v2f*)&xrow[kbase];                       // global_load_b64
        const float* wp = &Ws[(size_t)p * 2 * HID];
        v2f b0 = *(const v2f*)&wp[(rowl +  0) * 2];              // ds_load_b64
        v2f b1 = *(const v2f*)&wp[(rowl + 16) * 2];
        v2f b2 = *(const v2f*)&wp[(rowl + 32) * 2];
        v2f b3 = *(const v2f*)&wp[(rowl + 48) * 2];
        acc0 = __builtin_amdgcn_wmma_f32_16x16x4_f32(false, a, false, b0, (short)0, acc0, false, false);
        acc1 = __builtin_amdgcn_wmma_f32_16x16x4_f32(false, a, false, b1, (short)0, acc1, false, false);
        acc2 = __builtin_amdgcn_wmma_f32_16x16x4_f32(false, a, false, b2, (short)0, acc2, false, false);
        acc3 = __builtin_amdgcn_wmma_f32_16x16x4_f32(false, a, false, b3, (short)0, acc3, false, false);
    }

    float* out = h1 + (size_t)base * HID;
    for (int j = 0; j < 8; ++j) {
        size_t r = (size_t)(j + 8 * kh) * HID + rowl;
        out[r +  0] = acc0[j];
        out[r + 16] = acc1[j];
        out[r + 32] = acc2[j];
        out[r + 48] = acc3[j];
    }
}

// ---------------- GEMM2: h2[NN,8] = act1[NN,64] @ W2[64,8] (padded to 64x16) -------
// Same pair-interleaved LDS layout: Ws[p*32 + col*2 + t] = W2pad[(2p+t)*16 + col]
__global__ void gemm2_wmma_kernel(const float* __restrict__ act1, const float* __restrict__ W2,
                                  float* __restrict__ h2) {
    __shared__ float Ws[HID * 16];       // 4 KB, cols 8..15 zero
    for (int i = threadIdx.x; i < HID * 16; i += blockDim.x) {
        int p   = i >> 5;          // K-pair index (0..31)
        int rem = i & 31;
        int col = rem >> 1;        // 0..15
        int t   = rem & 1;
        Ws[i] = (col < C2) ? W2[(size_t)(2 * p + t) * C2 + col] : 0.0f;
    }
    __syncthreads();

    int wave = threadIdx.x >> 5;
    int lane = threadIdx.x & 31;
    int tile = blockIdx.x * (blockDim.x >> 5) + wave;
    int base = tile * 16;
    if (base >= NN) return;

    int rowl = lane & 15;
    int kh   = lane >> 4;
    const float* arow = act1 + (size_t)(base + rowl) * HID;

    v8f acc = {};
    for (int kk = 0; kk < HID; kk += 4) {
        int kbase = kk + 2 * kh;
        int p = kbase >> 1;
        v2f a = *(const v2f*)&arow[kbase];
        v2f b = *(const v2f*)&Ws[(size_t)p * 32 + rowl * 2];
        acc = __builtin_amdgcn_wmma_f32_16x16x4_f32(false, a, false, b, (short)0, acc, false, false);
    }

    if (rowl < C2) {
        for (int j = 0; j < 8; ++j)
            h2[(size_t)(base + j + 8 * kh) * C2 + rowl] = acc[j];
    }
}

// ---------------- edge aggregation, 64 channels: 16 threads per edge ----------------
__global__ void agg64_kernel(const int* __restrict__ src, const int* __restrict__ dst,
                             const float* __restrict__ dinv, const float* __restrict__ h,
                             float* __restrict__ out) {
    int idx = blockIdx.x * blockDim.x + threadIdx.x;
    int e = idx >> 4;
    if (e >= NE) return;
    // stream-ahead prefetch of the index arrays (global_prefetch_b8)
    __builtin_prefetch(src + e + 4096, 0, 0);
    __builtin_prefetch(dst + e + 4096, 0, 0);
    int c = (idx & 15) * 4;
    int s = src[e], d = dst[e];
    float w = dinv[s] * dinv[d];
    const float4 v = *(const float4*)(h + (size_t)s * HID + c);
    float* o = out + (size_t)d * HID + c;
    unsafeAtomicAdd(o + 0, v.x * w);
    unsafeAtomicAdd(o + 1, v.y * w);
    unsafeAtomicAdd(o + 2, v.z * w);
    unsafeAtomicAdd(o + 3, v.w * w);
}

// ---------------- edge aggregation, 8 channels: 2 threads per edge ----------------
__global__ void agg8_kernel(const int* __restrict__ src, const int* __restrict__ dst,
                            const float* __restrict__ dinv, const float* __restrict__ h,
                            float* __restrict__ out) {
    int idx = blockIdx.x * blockDim.x + threadIdx.x;
    int e = idx >> 1;
    if (e >= NE) return;
    __builtin_prefetch(src + e + 4096, 0, 0);
    __builtin_prefetch(dst + e + 4096, 0, 0);
    int c = (idx & 1) * 4;
    int s = src[e], d = dst[e];
    float w = dinv[s] * dinv[d];
    const float4 v = *(const float4*)(h + (size_t)s * C2 + c);
    float* o = out + (size_t)d * C2 + c;
    unsafeAtomicAdd(o + 0, v.x * w);
    unsafeAtomicAdd(o + 1, v.y * w);
    unsafeAtomicAdd(o + 2, v.z * w);
    unsafeAtomicAdd(o + 3, v.w * w);
}

// ---------------- self loop + bias + leaky (layer 1), in place on out1 -------------
__global__ void finish1_kernel(float* __restrict__ out1, const float* __restrict__ h1,
                               const float* __restrict__ dinv, const float* __restrict__ b1) {
    int idx = blockIdx.x * blockDim.x + threadIdx.x;
    int n = idx >> 4;
    if (n >= NN) return;
    int c = (idx & 15) * 4;
    float di = dinv[n];
    float w = di * di;
    float4 o = *(float4*)(out1 + (size_t)n * HID + c);
    float4 hv = *(const float4*)(h1 + (size_t)n * HID + c);
    o.x = lrelu(o.x + hv.x * w + b1[c + 0]);
    o.y = lrelu(o.y + hv.y * w + b1[c + 1]);
    o.z = lrelu(o.z + hv.z * w + b1[c + 2]);
    o.w = lrelu(o.w + hv.w * w + b1[c + 3]);
    *(float4*)(out1 + (size_t)n * HID + c) = o;
}

// ---------------- layer-2 finish fused with mean-pool accumulation ----------------
// pool layout: pool[0..511] = per-graph channel sums (NG*C2), pool[512..575] = counts
__global__ void finish2_pool_kernel(const float* __restrict__ out2, const float* __restrict__ h2,
                                    const float* __restrict__ dinv, const float* __restrict__ b2,
                                    const int* __restrict__ batch, float* __restrict__ pool) {
    __shared__ float lsum[NG * C2];
    __shared__ float lcnt[NG];
    int tid = threadIdx.x;
    for (int i = tid; i < NG * C2; i += blockDim.x) lsum[i] = 0.0f;
    if (tid < NG) lcnt[tid] = 0.0f;
    __syncthreads();

    int n = blockIdx.x * blockDim.x + tid;
    if (n < NN) {
        float di = dinv[n];
        float w = di * di;
        int g = batch[n];
        float vals[C2];
        #pragma unroll
        for (int c = 0; c < C2; ++c)
            vals[c] = lrelu(out2[(size_t)n * C2 + c] + h2[(size_t)n * C2 + c] * w + b2[c]);
        #pragma unroll
        for (int c = 0; c < C2; ++c) atomicAdd(&lsum[g * C2 + c], vals[c]);  // ds_add_f32
        atomicAdd(&lcnt[g], 1.0f);
    }
    __syncthreads();
    for (int i = tid; i < NG * C2; i += blockDim.x) unsafeAtomicAdd(&pool[i], lsum[i]);
    if (tid < NG) unsafeAtomicAdd(&pool[NG * C2 + tid], lcnt[tid]);
}

// ---------------- FC head: 64 graphs, one thread each ----------------
__global__ void final_fc_kernel(const float* __restrict__ pool, const float* __restrict__ gf,
                                const float* __restrict__ fcW1, const float* __restrict__ fcb1,
                                const float* __restrict__ fcW2, const float* __restrict__ fcb2,
                                float* __restrict__ out) {
    int g = blockIdx.x * blockDim.x + threadIdx.x;
    if (g >= NG) return;
    float z[GDIM + C2];
    float cnt = fmaxf(pool[NG * C2 + g], 1.0f);
    #pragma unroll
    for (int c = 0; c < C2; ++c) z[c] = pool[g * C2 + c] / cnt;
    #pragma unroll
    for (int c = 0; c < GDIM; ++c) z[C2 + c] = gf[g * GDIM + c];
    float acc2 = fcb2[0];
    #pragma unroll
    for (int j = 0; j < 16; ++j) {
        float t = fcb1[j];
        #pragma unroll
        for (int k = 0; k < GDIM + C2; ++k) t += z[k] * fcW1[k * 16 + j];
        acc2 += lrelu(t) * fcW2[j];
    }
    out[g] = lrelu(acc2);
}

// ---------------- host launcher ----------------
extern "C" void kernel_launch(void* const* d_in, const int* in_sizes, int n_in,
                              void* d_out, int out_size, void* d_ws, size_t ws_size,
                              hipStream_t stream) {
    const float* x     = (const float*)d_in[0];
    const int*   ei    = (const int*)d_in[1];
    const int*   src   = ei;
    const int*   dst   = ei + NE;
    const int*   batch = (const int*)d_in[2];
    const float* gf    = (const float*)d_in[3];
    const float* W1    = (const float*)d_in[4];
    const float* b1    = (const float*)d_in[5];
    const float* W2    = (const float*)d_in[6];
    const float* b2    = (const float*)d_in[7];
    const float* fcW1  = (const float*)d_in[8];
    const float* fcb1  = (const float*)d_in[9];
    const float* fcW2  = (const float*)d_in[10];
    const float* fcb2  = (const float*)d_in[11];
    float* out = (float*)d_out;

    float* ws = (float*)d_ws;
    size_t off = 0;
    auto carve = [&](size_t nflt) { float* p = ws + off; off += (nflt + 63) & ~(size_t)63; return p; };
    float* deg  = carve(NN);                  // becomes dinv in place
    float* h1   = carve((size_t)NN * HID);
    float* o1   = carve((size_t)NN * HID);    // becomes act1 in place
    float* h2   = carve((size_t)NN * C2);
    float* o2   = carve((size_t)NN * C2);
    float* pool = carve(NG * C2 + NG);

    const int B = 256;
    // init
    fill_kernel<<<(NN + B - 1) / B, B, 0, stream>>>(deg, 1.0f, NN);
    fill_kernel<<<((size_t)NN * HID + B - 1) / B, B, 0, stream>>>(o1, 0.0f, NN * HID);
    fill_kernel<<<((size_t)NN * C2 + B - 1) / B, B, 0, stream>>>(o2, 0.0f, NN * C2);
    fill_kernel<<<1, 576, 0, stream>>>(pool, 0.0f, NG * C2 + NG);

    // degrees -> dinv
    deg_accum_kernel<<<(NE + B - 1) / B, B, 0, stream>>>(dst, deg, NE);
    dinv_kernel<<<(NN + B - 1) / B, B, 0, stream>>>(deg, NN);

    // layer 1
    int tiles = (NN + 15) / 16;               // 6250
    gemm1_wmma_kernel<<<(tiles + 3) / 4, 128, 0, stream>>>(x, W1, h1);
    agg64_kernel<<<((size_t)NE * 16 + B - 1) / B, B, 0, stream>>>(src, dst, deg, h1, o1);
    finish1_kernel<<<((size_t)NN * 16 + B - 1) / B, B, 0, stream>>>(o1, h1, deg, b1);

    // layer 2
    gemm2_wmma_kernel<<<(tiles + 3) / 4, 128, 0, stream>>>(o1, W2, h2);
    agg8_kernel<<<((size_t)NE * 2 + B - 1) / B, B, 0, stream>>>(src, dst, deg, h2, o2);
    finish2_pool_kernel<<<(NN + B - 1) / B, B, 0, stream>>>(o2, h2, deg, b2, batch, pool);

    // head
    final_fc_kernel<<<1, NG, 0, stream>>>(pool, gf, fcW1, fcb1, fcW2, fcb2, out);
}